// SRNNModel_63093069578506
// MI455X (gfx1250) — compile-verified
//
#include <hip/hip_runtime.h>
#include <math.h>

#define HID 512
#define NCLS 50000

typedef float v2f __attribute__((ext_vector_type(2)));
typedef float v8f __attribute__((ext_vector_type(8)));

__device__ __forceinline__ float sigmoidf(float x) {
    return 1.0f / (1.0f + __expf(-x));
}

// GEMV tile via V_WMMA_F32_16X16X4_F32.
// Computes y[m] = dot(W[row0+m, 0:K], v_lds[0:K]) for m = 0..15.
// A (16x4 f32) layout per ISA: lanes 0-15 hold A[m, {0,1}], lanes 16-31 hold A[m, {2,3}].
// B (4x16) is the vector broadcast across all 16 columns, so every D column equals y.
// Result: lanes 0-15 hold y[0..7] in acc[0..7]; lanes 16-31 hold y[8..15].
__device__ __forceinline__ v8f wmma_gemv_tile(const float* __restrict__ W,
                                              long row0, int K,
                                              const float* v_lds) {
    const int lane = threadIdx.x & 31;
    const int half = lane >> 4;            // 0: k pair {0,1}; 1: k pair {2,3}
    const int m    = lane & 15;
    const float* wrow = W + (long)(row0 + m) * K + half * 2;
    v8f acc = {0.f, 0.f, 0.f, 0.f, 0.f, 0.f, 0.f, 0.f};
    for (int k = 0; k < K; k += 4) {
        v2f a = *(const v2f*)(wrow + k);                   // global_load_b64, 8B aligned
        v2f b = *(const v2f*)(v_lds + k + half * 2);       // ds_load_b64 broadcast
        acc = __builtin_amdgcn_wmma_f32_16x16x4_f32(
            /*neg_a=*/false, a, /*neg_b=*/false, b,
            /*c_mod=*/(short)0, acc, /*reuse_a=*/false, /*reuse_b=*/false);
    }
    return acc;
}

// ---- Kernel A: h3_new = sigmoid(interval*W_int + b_int + h3 @ W_h3.T + b_h3) ----
// 32 tiles of 16 rows; 1 wave per tile; 4 blocks x 256 threads.
__global__ void k_h3(const float* __restrict__ h3, const float* __restrict__ W_h3,
                     const float* __restrict__ b_h3, const float* __restrict__ W_int,
                     const float* __restrict__ b_int, const float* __restrict__ intervalX,
                     float* __restrict__ h3_new) {
    __shared__ float v[HID];
    for (int i = threadIdx.x; i < HID; i += blockDim.x) v[i] = h3[i];
    __syncthreads();
    const int tile = blockIdx.x * (blockDim.x >> 5) + (threadIdx.x >> 5);
    const long row0 = (long)tile * 16;
    v8f acc = wmma_gemv_tile(W_h3, row0, HID, v);
    const int lane = threadIdx.x & 31;
    const float iv = intervalX[0];
    if (lane == 0 || lane == 16) {
        const int mbase = (int)row0 + (lane >> 4) * 8;
#pragma unroll
        for (int t = 0; t < 8; ++t) {
            const int r = mbase + t;
            h3_new[r] = sigmoidf(acc[t] + iv * W_int[r] + b_int[r] + b_h3[r]);
        }
    }
}

// ---- Kernel B (dominant, 512 MB stream): hpre[j] = sum_i h[i]*sigmoid(W_ctx[j*H+i,:]@h3_new + b_ctx)
// One block per output j (512 blocks x 256 threads = 8 waves). Each wave handles 4 tiles
// of 16 rows; rows for a fixed j never cross blocks, so W_ctx is read exactly once.
__global__ void k_ctx(const float* __restrict__ h3_new, const float* __restrict__ h,
                      const float* __restrict__ W_ctx, const float* __restrict__ b_ctx,
                      float* __restrict__ hpre) {
    __shared__ float v[HID];
    __shared__ float hv[HID];
    __shared__ float wsum[8];
    for (int i = threadIdx.x; i < HID; i += blockDim.x) { v[i] = h3_new[i]; hv[i] = h[i]; }
    __syncthreads();
    const int j    = blockIdx.x;
    const int wave = threadIdx.x >> 5;
    const int lane = threadIdx.x & 31;
    const int half = lane >> 4;
    float partial = 0.0f;
    for (int t = wave; t < 32; t += 8) {
        const int i0 = t * 16;
        const long row0 = (long)j * HID + i0;
        v8f acc = wmma_gemv_tile(W_ctx, row0, HID, v);
        // lanes 0-15: rows i0..i0+7 ; lanes 16-31: rows i0+8..i0+15 (all redundant within half)
        const int ibase = i0 + half * 8;
        float s = 0.0f;
#pragma unroll
        for (int tt = 0; tt < 8; ++tt) {
            const int i = ibase + tt;
            s += hv[i] * sigmoidf(acc[tt] + b_ctx[(long)j * HID + i]);
        }
        partial += s;
    }
    // one representative lane per half-wave
    const float tot = __shfl(partial, 0, 32) + __shfl(partial, 16, 32);
    if (lane == 0) wsum[wave] = tot;
    __syncthreads();
    if (threadIdx.x == 0) {
        float r = 0.0f;
#pragma unroll
        for (int w = 0; w < 8; ++w) r += wsum[w];
        hpre[j] = r;
    }
}

// ---- Kernel C: h_new = sigmoid(emb[inputX] @ Wx.T + Bx + hpre + b_ictx) ----
__global__ void k_h(const int* __restrict__ inputX, const float* __restrict__ emb,
                    const float* __restrict__ Wx, const float* __restrict__ Bx,
                    const float* __restrict__ b_ictx, const float* __restrict__ hpre,
                    float* __restrict__ h_new) {
    __shared__ float v[HID];
    const float* xe = emb + (long)inputX[0] * HID;
    for (int i = threadIdx.x; i < HID; i += blockDim.x) v[i] = xe[i];
    __syncthreads();
    const int tile = blockIdx.x * (blockDim.x >> 5) + (threadIdx.x >> 5);
    const long row0 = (long)tile * 16;
    v8f acc = wmma_gemv_tile(Wx, row0, HID, v);
    const int lane = threadIdx.x & 31;
    if (lane == 0 || lane == 16) {
        const int mbase = (int)row0 + (lane >> 4) * 8;
#pragma unroll
        for (int t = 0; t < 8; ++t) {
            const int r = mbase + t;
            h_new[r] = sigmoidf(acc[t] + Bx[r] + b_ictx[r] + hpre[r]);
        }
    }
}

// ---- Kernel D (102 MB stream): logits = h_new @ fc_W.T + fc_b ----
// 50000 = 3125 tiles of 16, 1 wave each; 391 blocks x 256 threads, guard last.
__global__ void k_logits(const float* __restrict__ h_new, const float* __restrict__ fc_W,
                         const float* __restrict__ fc_b, float* __restrict__ logits) {
    __shared__ float v[HID];
    for (int i = threadIdx.x; i < HID; i += blockDim.x) v[i] = h_new[i];
    __syncthreads();
    const int tile = blockIdx.x * (blockDim.x >> 5) + (threadIdx.x >> 5);
    if (tile >= NCLS / 16) return;                // wave-uniform exit; EXEC stays all-1s
    const long row0 = (long)tile * 16;
    v8f acc = wmma_gemv_tile(fc_W, row0, HID, v);
    const int lane = threadIdx.x & 31;
    if (lane == 0 || lane == 16) {
        const int mbase = (int)row0 + (lane >> 4) * 8;
#pragma unroll
        for (int t = 0; t < 8; ++t) {
            const int r = mbase + t;
            logits[r] = acc[t] + fc_b[r];
        }
    }
}

extern "C" void kernel_launch(void* const* d_in, const int* in_sizes, int n_in,
                              void* d_out, int out_size, void* d_ws, size_t ws_size,
                              hipStream_t stream) {
    // setup_inputs order:
    const int*   inputX    = (const int*)  d_in[0];
    // d_in[1] hourX, d_in[2] weekdayX: unused by the reference computation
    const float* intervalX = (const float*)d_in[3];
    const float* h         = (const float*)d_in[4];
    const float* h3        = (const float*)d_in[5];
    const float* emb       = (const float*)d_in[6];
    const float* W_h3      = (const float*)d_in[7];
    const float* b_h3      = (const float*)d_in[8];
    const float* W_int     = (const float*)d_in[9];
    const float* b_int     = (const float*)d_in[10];
    const float* W_ctx     = (const float*)d_in[11];
    const float* b_ctx     = (const float*)d_in[12];
    const float* b_ictx    = (const float*)d_in[13];
    const float* Wx        = (const float*)d_in[14];
    const float* Bx        = (const float*)d_in[15];
    const float* fc_W      = (const float*)d_in[16];
    const float* fc_b      = (const float*)d_in[17];

    float* logits = (float*)d_out;               // [50000]
    float* h_new  = logits + NCLS;               // [512]
    float* h3_new = h_new + HID;                 // [512]
    float* hpre   = (float*)d_ws;                // [512] scratch: h @ ctx.T

    k_h3   <<<4,   256, 0, stream>>>(h3, W_h3, b_h3, W_int, b_int, intervalX, h3_new);
    k_ctx  <<<HID, 256, 0, stream>>>(h3_new, h, W_ctx, b_ctx, hpre);
    k_h    <<<4,   256, 0, stream>>>(inputX, emb, Wx, Bx, b_ictx, hpre, h_new);
    k_logits<<<(NCLS / 16 + 7) / 8, 256, 0, stream>>>(h_new, fc_W, fc_b, logits);
}